// GraphSAGEModel_34600256537252
// MI455X (gfx1250) — compile-verified
//
#include <hip/hip_runtime.h>
#include <hip/hip_bf16.h>

typedef float v2f __attribute__((ext_vector_type(2)));
typedef float v8f __attribute__((ext_vector_type(8)));

#define D 128

// ---------------------------------------------------------------- utilities
__global__ __launch_bounds__(256) void fill_zero_kernel(float* __restrict__ p, int n) {
    int i = blockIdx.x * blockDim.x + threadIdx.x;
    int stride = gridDim.x * blockDim.x;
    for (; i < n; i += stride) p[i] = 0.0f;
}

// deg[dst] += 1 per edge (fp32 so the combine kernel can consume directly)
__global__ __launch_bounds__(256) void degree_kernel(const int* __restrict__ dst,
                                                     float* __restrict__ deg, int E) {
    int e = blockIdx.x * blockDim.x + threadIdx.x;
    if (e < E) atomicAdd(&deg[dst[e]], 1.0f);
}

// one wave per edge: 32 lanes x float4 = 128 features; gather hits L2 (x fits
// in 192MB L2), scatter-add via global_atomic_add_f32
__global__ __launch_bounds__(256) void scatter_kernel(const float* __restrict__ x,
                                                      const int* __restrict__ src,
                                                      const int* __restrict__ dst,
                                                      float* __restrict__ agg, int E) {
    int gid = blockIdx.x * blockDim.x + threadIdx.x;
    int e = gid >> 5;
    int lane = gid & 31;
    if (e >= E) return;
    int s = src[e];
    int d = dst[e];
    const float4* xs = (const float4*)(x + (size_t)s * D);
    float4 v = xs[lane];
    float* ad = agg + (size_t)d * D + lane * 4;
    atomicAdd(ad + 0, v.x);
    atomicAdd(ad + 1, v.y);
    atomicAdd(ad + 2, v.z);
    atomicAdd(ad + 3, v.w);
}

// ------------------------------------------------- fused SAGEConv via WMMA f32
// out = relu( (aggsum/deg) @ Wl + b + xin @ Wr )
// Block = 256 threads = 8 waves; block handles 16 rows x 128 cols.
// Wave w computes the 16x16 tile at columns [16w, 16w+16); K-loop of 32
// V_WMMA_F32_16X16X4_F32 steps per GEMM, two GEMMs share one accumulator.
__global__ __launch_bounds__(256) void sage_combine_kernel(
    const float* __restrict__ aggsum, const float* __restrict__ deg,
    const float* __restrict__ xin,
    const float* __restrict__ Wl, const float* __restrict__ bias,
    const float* __restrict__ Wr, float* __restrict__ out) {
    const int lane   = threadIdx.x & 31;
    const int wave   = threadIdx.x >> 5;       // 0..7  -> column tile
    const int rowblk = blockIdx.x;             // N/16 blocks, N % 16 == 0
    const int r      = lane & 15;              // row within tile (A/C halves)
    const int hi     = lane >> 4;              // 0 / 1
    const int khalf  = hi * 2;                 // A/B k-pair select per lane
    const int col    = wave * 16 + r;          // output column for B/C/D

    const int rowA = rowblk * 16 + r;
    const float invd = 1.0f / fmaxf(deg[rowA], 1.0f);   // mean = sum * invd

    const float* Aagg = aggsum + (size_t)rowA * D;
    const float* Ax   = xin    + (size_t)rowA * D;

    v8f acc = {};
    #pragma unroll
    for (int k = 0; k < D; k += 4) {
        const int ka = k + khalf;
        // A fragment: 16x4 fp32, mean-scaled on the fly
        v2f a_agg = { Aagg[ka] * invd, Aagg[ka + 1] * invd };
        v2f a_x   = { Ax[ka],          Ax[ka + 1] };
        // B fragment: 4x16 fp32 from row-major [D, D] weight
        v2f b_l = { Wl[(size_t)ka * D + col], Wl[(size_t)(ka + 1) * D + col] };
        v2f b_r = { Wr[(size_t)ka * D + col], Wr[(size_t)(ka + 1) * D + col] };
        acc = __builtin_amdgcn_wmma_f32_16x16x4_f32(false, a_agg, false, b_l,
                                                    (short)0, acc, false, false);
        acc = __builtin_amdgcn_wmma_f32_16x16x4_f32(false, a_x, false, b_r,
                                                    (short)0, acc, false, false);
    }

    const float bcol = bias[col];
    #pragma unroll
    for (int v = 0; v < 8; ++v) {
        float val = acc[v] + bcol;
        val = val > 0.0f ? val : 0.0f;                  // ReLU
        const int rowO = rowblk * 16 + v + 8 * hi;      // C/D VGPR layout
        out[(size_t)rowO * D + col] = val;
    }
}

// --------------------------------------------------- final projection to D_OUT=1
__global__ __launch_bounds__(256) void final_kernel(const float* __restrict__ h,
                                                    const float* __restrict__ Wo,
                                                    const float* __restrict__ bo,
                                                    float* __restrict__ out, int N) {
    int i = blockIdx.x * blockDim.x + threadIdx.x;
    if (i >= N) return;
    const float4* hp = (const float4*)(h + (size_t)i * D);
    const float4* wp = (const float4*)Wo;   // Wo is [128,1] -> contiguous 128 floats
    float s = 0.0f;
    #pragma unroll
    for (int j = 0; j < D / 4; ++j) {
        float4 v = hp[j];
        float4 w = wp[j];
        s += v.x * w.x + v.y * w.y + v.z * w.z + v.w * w.w;
    }
    out[i] = s + bo[0];
}

// ---------------------------------------------------------------- launcher
extern "C" void kernel_launch(void* const* d_in, const int* in_sizes, int n_in,
                              void* d_out, int out_size, void* d_ws, size_t ws_size,
                              hipStream_t stream) {
    const float* x   = (const float*)d_in[0];
    const int*  edge = (const int*)d_in[1];
    const float* W1l = (const float*)d_in[2];
    const float* b1  = (const float*)d_in[3];
    const float* W1r = (const float*)d_in[4];
    const float* W2l = (const float*)d_in[5];
    const float* b2  = (const float*)d_in[6];
    const float* W2r = (const float*)d_in[7];
    const float* Wo  = (const float*)d_in[8];
    const float* bo  = (const float*)d_in[9];
    float* out = (float*)d_out;

    const int N = in_sizes[0] / D;      // 40000
    const int E = in_sizes[1] / 2;      // 640000
    const int* src = edge;              // edge_index[0, :]
    const int* dst = edge + E;          // edge_index[1, :]

    float* ws  = (float*)d_ws;
    const size_t nd = (size_t)N * D;
    float* agg = ws;                    // [N, D] scatter-add accumulator
    float* h1  = ws + nd;               // [N, D]
    float* h2  = ws + 2 * nd;           // [N, D]
    float* deg = ws + 3 * nd;           // [N]

    const int edge_threads = E * 32;

    // ---- degree (graph is identical for both layers; compute once)
    fill_zero_kernel<<<1024, 256, 0, stream>>>(deg, N);
    degree_kernel<<<(E + 255) / 256, 256, 0, stream>>>(dst, deg, E);

    // ---- layer 1
    fill_zero_kernel<<<4096, 256, 0, stream>>>(agg, (int)nd);
    scatter_kernel<<<(edge_threads + 255) / 256, 256, 0, stream>>>(x, src, dst, agg, E);
    sage_combine_kernel<<<N / 16, 256, 0, stream>>>(agg, deg, x, W1l, b1, W1r, h1);

    // ---- layer 2
    fill_zero_kernel<<<4096, 256, 0, stream>>>(agg, (int)nd);
    scatter_kernel<<<(edge_threads + 255) / 256, 256, 0, stream>>>(h1, src, dst, agg, E);
    sage_combine_kernel<<<N / 16, 256, 0, stream>>>(agg, deg, h1, W2l, b2, W2r, h2);

    // ---- output head
    final_kernel<<<(N + 255) / 256, 256, 0, stream>>>(h2, Wo, bo, out, N);
}